// EdgePolicyModel_65017214926934
// MI455X (gfx1250) — compile-verified
//
#include <hip/hip_runtime.h>

typedef __attribute__((ext_vector_type(2))) float v2f;
typedef __attribute__((ext_vector_type(8))) float v8f;

#define GG 256  // number of graphs (fixed by reference)

// ---------------------------------------------------------------- degree
__global__ void k_degree(const int* __restrict__ row, float* __restrict__ deg, int E) {
    int i = blockIdx.x * blockDim.x + threadIdx.x;
    if (i < E) atomicAdd(&deg[row[i]], 1.0f);
}

__global__ void k_dis(const float* __restrict__ deg, float* __restrict__ dis, int n) {
    int i = blockIdx.x * blockDim.x + threadIdx.x;
    if (i < n) {
        float d = deg[i];
        dis[i] = (d > 0.0f) ? rsqrtf(fmaxf(d, 1.0f)) : 0.0f;
    }
}

// ------------------------------------------------- edge scatter, layer 1 (F=8)
__global__ void k_scatter1(const int* __restrict__ row, const int* __restrict__ col,
                           const float* __restrict__ dis, const float* __restrict__ x,
                           float* __restrict__ tx1, int E) {
    int i = blockIdx.x * blockDim.x + threadIdx.x;
    if (i >= E) return;
    int r = row[i], c = col[i];
    float w = -(dis[r] * dis[c]);
    const float4* xr = reinterpret_cast<const float4*>(x) + (size_t)r * 2;
    float4 a = xr[0], b = xr[1];
    float* dst = tx1 + (size_t)c * 8;
    atomicAdd(dst + 0, w * a.x); atomicAdd(dst + 1, w * a.y);
    atomicAdd(dst + 2, w * a.z); atomicAdd(dst + 3, w * a.w);
    atomicAdd(dst + 4, w * b.x); atomicAdd(dst + 5, w * b.y);
    atomicAdd(dst + 6, w * b.z); atomicAdd(dst + 7, w * b.w);
}

// ------------------------------------------------- edge scatter, layer 2 (C=16)
__global__ void k_scatter2(const int* __restrict__ row, const int* __restrict__ col,
                           const float* __restrict__ dis, const float* __restrict__ h1,
                           float* __restrict__ tx2, int E) {
    int i = blockIdx.x * blockDim.x + threadIdx.x;
    if (i >= E) return;
    int r = row[i], c = col[i];
    float w = -(dis[r] * dis[c]);
    const float4* hr = reinterpret_cast<const float4*>(h1) + (size_t)r * 4;
    float* dst = tx2 + (size_t)c * 16;
#pragma unroll
    for (int q = 0; q < 4; ++q) {
        float4 v = hr[q];
        atomicAdd(dst + q * 4 + 0, w * v.x);
        atomicAdd(dst + q * 4 + 1, w * v.y);
        atomicAdd(dst + q * 4 + 2, w * v.z);
        atomicAdd(dst + q * 4 + 3, w * v.w);
    }
}

// ---------------------------------------------------------------- WMMA helpers
// A-matrix 16x4 f32 (ISA 7.12.2): lane<16 -> row=lane, holds K={k0,k0+1};
// lane>=16 -> row=lane-16, holds K={k0+2,k0+3}.  src row-major [node][ld].
__device__ inline v2f load_A_f32(const float* __restrict__ src, int node0,
                                 int ld, int kbase, int l16, int half) {
    const float* p = src + (size_t)(node0 + l16) * ld + kbase + 2 * half;
    v2f a; a.x = p[0]; a.y = p[1];
    return a;
}
// B-matrix 4x16 f32, row-major weights W[k][n] (ld=16): v0 lanes0-15 = K=k0 row,
// lanes16-31 = K=k0+2 row; v1 = K=k0+1 / K=k0+3.
__device__ inline v2f load_B_f32(const float* __restrict__ W, int kbase,
                                 int l16, int half) {
    const float* p = W + (size_t)(kbase + 2 * half) * 16 + l16;
    v2f b; b.x = p[0]; b.y = p[16];
    return b;
}

// ---------------------------------------- layer 1: h1 = relu(x@W0 + tx1@W1 + b)
__global__ __launch_bounds__(256)
void k_cheb1(const float* __restrict__ x, const float* __restrict__ tx1,
             const float* __restrict__ W0, const float* __restrict__ W1,
             const float* __restrict__ b, float* __restrict__ h1, int ntiles) {
    int wave = (blockIdx.x * blockDim.x + threadIdx.x) >> 5;
    if (wave >= ntiles) return;          // whole-wave uniform: EXEC all 1s inside
    int lane = threadIdx.x & 31;
    int half = lane >> 4, l16 = lane & 15;
    int node0 = wave * 16;

    v8f acc = {};
#pragma unroll
    for (int kc = 0; kc < 2; ++kc) {     // K = 8 -> two 16x16x4 steps
        v2f a = load_A_f32(x, node0, 8, kc * 4, l16, half);
        v2f w = load_B_f32(W0, kc * 4, l16, half);
        acc = __builtin_amdgcn_wmma_f32_16x16x4_f32(false, a, false, w,
                                                    (short)0, acc, false, false);
    }
#pragma unroll
    for (int kc = 0; kc < 2; ++kc) {
        v2f a = load_A_f32(tx1, node0, 8, kc * 4, l16, half);
        v2f w = load_B_f32(W1, kc * 4, l16, half);
        acc = __builtin_amdgcn_wmma_f32_16x16x4_f32(false, a, false, w,
                                                    (short)0, acc, false, false);
    }
    float bias = b[l16];
#pragma unroll
    for (int i = 0; i < 8; ++i) {        // C/D layout: VGPR i -> M=i (+8 for hi half)
        int node = node0 + i + half * 8;
        float v = fmaxf(acc[i] + bias, 0.0f);
        h1[(size_t)node * 16 + l16] = v;
    }
}

// ---- layer 2 + readout: s = relu(h1@W0 + tx2@W1 + b) @ Wr + br   (per node)
__global__ __launch_bounds__(256)
void k_cheb2(const float* __restrict__ h1, const float* __restrict__ tx2,
             const float* __restrict__ W0, const float* __restrict__ W1,
             const float* __restrict__ b, const float* __restrict__ Wr,
             const float* __restrict__ br, float* __restrict__ s, int ntiles) {
    int wave = (blockIdx.x * blockDim.x + threadIdx.x) >> 5;
    if (wave >= ntiles) return;
    int lane = threadIdx.x & 31;
    int half = lane >> 4, l16 = lane & 15;
    int node0 = wave * 16;

    v8f acc = {};
#pragma unroll
    for (int kc = 0; kc < 4; ++kc) {     // K = 16 -> four 16x16x4 steps
        v2f a = load_A_f32(h1, node0, 16, kc * 4, l16, half);
        v2f w = load_B_f32(W0, kc * 4, l16, half);
        acc = __builtin_amdgcn_wmma_f32_16x16x4_f32(false, a, false, w,
                                                    (short)0, acc, false, false);
    }
#pragma unroll
    for (int kc = 0; kc < 4; ++kc) {
        v2f a = load_A_f32(tx2, node0, 16, kc * 4, l16, half);
        v2f w = load_B_f32(W1, kc * 4, l16, half);
        acc = __builtin_amdgcn_wmma_f32_16x16x4_f32(false, a, false, w,
                                                    (short)0, acc, false, false);
    }
    float bias = b[l16];
    float wr = Wr[l16];
    float brv = br[0];
#pragma unroll
    for (int i = 0; i < 8; ++i) {
        float v = fmaxf(acc[i] + bias, 0.0f) * wr;   // partial of h2[node]·Wr
        // reduce over the 16 channels (lanes within each half of the wave)
        v += __shfl_xor(v, 1, 32);
        v += __shfl_xor(v, 2, 32);
        v += __shfl_xor(v, 4, 32);
        v += __shfl_xor(v, 8, 32);
        if (l16 == 0) s[node0 + i + half * 8] = v + brv;
    }
}

// ---------------------------------------------------------------- softmax
__device__ inline unsigned f2mono(float f) {
    unsigned u = __float_as_uint(f);
    return (u & 0x80000000u) ? ~u : (u | 0x80000000u);
}
__device__ inline float mono2f(unsigned u) {
    return __uint_as_float((u & 0x80000000u) ? (u & 0x7FFFFFFFu) : ~u);
}

__global__ void k_segmax(const float* __restrict__ s, const int* __restrict__ batch,
                         unsigned* __restrict__ menc, int n) {
    int i = blockIdx.x * blockDim.x + threadIdx.x;
    if (i < n) atomicMax(&menc[batch[i]], f2mono(s[i]));
}

__global__ void k_segexp(const float* __restrict__ s, const int* __restrict__ batch,
                         const unsigned* __restrict__ menc, float* __restrict__ z,
                         float* __restrict__ out, int n) {
    int i = blockIdx.x * blockDim.x + threadIdx.x;
    if (i < n) {
        float e = expf(s[i] - mono2f(menc[batch[i]]));
        out[i] = e;
        atomicAdd(&z[batch[i]], e);
    }
}

__global__ void k_segdiv(const int* __restrict__ batch, const float* __restrict__ z,
                         float* __restrict__ out, int n) {
    int i = blockIdx.x * blockDim.x + threadIdx.x;
    if (i < n) out[i] = out[i] / z[batch[i]];
}

// ---------------------------------------------------------------- launch
extern "C" void kernel_launch(void* const* d_in, const int* in_sizes, int n_in,
                              void* d_out, int out_size, void* d_ws, size_t ws_size,
                              hipStream_t stream) {
    const float* x    = (const float*)d_in[0];
    const int*   ei   = (const int*)d_in[1];
    const int*   batch= (const int*)d_in[2];
    const float* W1_0 = (const float*)d_in[3];
    const float* W1_1 = (const float*)d_in[4];
    const float* b1   = (const float*)d_in[5];
    const float* W2_0 = (const float*)d_in[6];
    const float* W2_1 = (const float*)d_in[7];
    const float* b2   = (const float*)d_in[8];
    const float* Wr   = (const float*)d_in[9];
    const float* br   = (const float*)d_in[10];

    const int N = in_sizes[0] / 8;       // 200,000
    const int E = in_sizes[1] / 2;       // 12,800,000
    const int* row = ei;
    const int* col = ei + E;

    // workspace layout (floats)
    float* deg = (float*)d_ws;                       // N
    float* dis = deg + N;                            // N
    float* tx1 = dis + N;                            // N*8
    float* h1  = tx1 + (size_t)N * 8;                // N*16
    float* tx2 = h1  + (size_t)N * 16;               // N*16
    float* sco = tx2 + (size_t)N * 16;               // N
    unsigned* menc = (unsigned*)(sco + N);           // GG
    float* z = (float*)(menc + GG);                  // GG

    // zero the accumulators every call (graph-replay safe)
    hipMemsetAsync(deg, 0, (size_t)N * sizeof(float), stream);
    hipMemsetAsync(tx1, 0, (size_t)N * 8 * sizeof(float), stream);
    hipMemsetAsync(tx2, 0, (size_t)N * 16 * sizeof(float), stream);
    hipMemsetAsync(menc, 0, (size_t)GG * (sizeof(unsigned) + sizeof(float)), stream);

    const int TB = 256;
    const int gE = (E + TB - 1) / TB;
    const int gN = (N + TB - 1) / TB;
    const int ntiles = N / 16;                       // 12,500 (exact)
    const int gT = (ntiles + 8 - 1) / 8;             // 8 waves per block

    k_degree <<<gE, TB, 0, stream>>>(row, deg, E);
    k_dis    <<<gN, TB, 0, stream>>>(deg, dis, N);
    k_scatter1<<<gE, TB, 0, stream>>>(row, col, dis, x, tx1, E);
    k_cheb1  <<<gT, TB, 0, stream>>>(x, tx1, W1_0, W1_1, b1, h1, ntiles);
    k_scatter2<<<gE, TB, 0, stream>>>(row, col, dis, h1, tx2, E);
    k_cheb2  <<<gT, TB, 0, stream>>>(h1, tx2, W2_0, W2_1, b2, Wr, br, sco, ntiles);
    k_segmax <<<gN, TB, 0, stream>>>(sco, batch, menc, N);
    k_segexp <<<gN, TB, 0, stream>>>(sco, batch, menc, z, (float*)d_out, N);
    k_segdiv <<<gN, TB, 0, stream>>>(batch, z, (float*)d_out, N);
}